// SIRSDESimulator_7584912245541
// MI455X (gfx1250) — compile-verified
//
#include <hip/hip_runtime.h>
#include <stdint.h>

#define TIME_STEPS 49
#define STEPS_PER_SAVE 100
#define STEPS_TOTAL (TIME_STEPS * STEPS_PER_SAVE)          // 4900
#define BATCH 8192
#define NOISE_N ((uint32_t)STEPS_TOTAL * (uint32_t)BATCH)  // 40,140,800
#define NHALF (NOISE_N / 2u)                               // 20,070,400
#define HALF_ST (STEPS_TOTAL / 2)                          // 2450
#define DTC 0.01f
#define SQRT_DT 0.1f

#define CHUNK_ROWS 20                                      // steps per LDS tile
#define NUM_CHUNKS (STEPS_TOTAL / CHUNK_ROWS)              // 245
#define CHUNKS_PER_SAVE (STEPS_PER_SAVE / CHUNK_ROWS)      // 5
#define TPB 256

// ---------------- threefry2x32 (exact JAX schedule) ----------------
__device__ __forceinline__ uint32_t rotl32(uint32_t x, uint32_t r) {
  return (x << r) | (x >> (32u - r));
}

__device__ __forceinline__ void tf2x32(uint32_t k0, uint32_t k1,
                                       uint32_t c0, uint32_t c1,
                                       uint32_t& o0, uint32_t& o1) {
  const uint32_t ks2 = k0 ^ k1 ^ 0x1BD11BDAu;
  uint32_t x0 = c0 + k0, x1 = c1 + k1;
#define TF_R(r) { x0 += x1; x1 = rotl32(x1, (r)); x1 ^= x0; }
  TF_R(13u) TF_R(15u) TF_R(26u) TF_R(6u)   x0 += k1;  x1 += ks2 + 1u;
  TF_R(17u) TF_R(29u) TF_R(16u) TF_R(24u)  x0 += ks2; x1 += k0  + 2u;
  TF_R(13u) TF_R(15u) TF_R(26u) TF_R(6u)   x0 += k0;  x1 += k1  + 3u;
  TF_R(17u) TF_R(29u) TF_R(16u) TF_R(24u)  x0 += k1;  x1 += ks2 + 4u;
  TF_R(13u) TF_R(15u) TF_R(26u) TF_R(6u)   x0 += ks2; x1 += k0  + 5u;
#undef TF_R
  o0 = x0; o1 = x1;
}

__device__ __forceinline__ float u01(uint32_t b) {
  return __uint_as_float((b >> 9) | 0x3F800000u) - 1.0f;
}

__device__ __forceinline__ float normal_scaled(uint32_t b) {
  const float lo = -0.99999994f;               // nextafter(-1, 0) in f32
  float u = u01(b) * (1.0f - lo) + lo;
  u = fmaxf(u, lo);
  return (1.41421356f * erfinvf(u)) * SQRT_DT;
}

// ---------------- CDNA5 async global->LDS helpers ----------------
__device__ __forceinline__ void async_load_b128(uint32_t lds_byte, const float* gaddr) {
  asm volatile("global_load_async_to_lds_b128 %0, %1, off"
               :: "v"(lds_byte), "v"(gaddr)
               : "memory");
}

__device__ __forceinline__ void wait_async_le5() {
#if __has_builtin(__builtin_amdgcn_s_wait_asynccnt)
  __builtin_amdgcn_s_wait_asynccnt(5);
#else
  asm volatile("s_wait_asynccnt 0x5" ::: "memory");
#endif
}

__device__ __forceinline__ void wait_async_le0() {
#if __has_builtin(__builtin_amdgcn_s_wait_asynccnt)
  __builtin_amdgcn_s_wait_asynccnt(0);
#else
  asm volatile("s_wait_asynccnt 0x0" ::: "memory");
#endif
}

// ---------------- key derivation: jax.random.split(key(seed)) ----------------
__global__ void derive_keys_kernel(const int* __restrict__ seedp,
                                   uint32_t* __restrict__ keys) {
  if (blockIdx.x == 0 && threadIdx.x == 0) {
    long long sd = (long long)(*seedp);
    uint32_t k0 = (uint32_t)(((unsigned long long)sd) >> 32);
    uint32_t k1 = (uint32_t)((unsigned long long)sd);
    uint32_t a0, a1, b0, b1;
    tf2x32(k0, k1, 0u, 2u, a0, a1);
    tf2x32(k0, k1, 1u, 3u, b0, b1);
    keys[0] = a0; keys[1] = b0;       // k_noise
    keys[2] = a1; keys[3] = b1;       // k_u
  }
}

// ---------------- phase 1: full-machine noise generation ----------------
__global__ void gen_noise_kernel(float* __restrict__ ws,
                                 const uint32_t* __restrict__ keys) {
  uint32_t j = blockIdx.x * blockDim.x + threadIdx.x;
  if (j >= NHALF) return;
  uint32_t kn0 = keys[0], kn1 = keys[1];
  uint32_t o0, o1;
  tf2x32(kn0, kn1, j, j + NHALF, o0, o1);
  ws[j]         = normal_scaled(o0);
  ws[j + NHALF] = normal_scaled(o1);
}

// ---------------- shared summary epilogue ----------------
__device__ __forceinline__ void write_summary(float* __restrict__ out, int b,
                                              uint32_t ku0, uint32_t ku1,
                                              float mx, int mi,
                                              float dsum, float dsq) {
  uint32_t o0, o1; float u;
  if (b < BATCH / 2) { tf2x32(ku0, ku1, (uint32_t)b, (uint32_t)b + BATCH / 2u, o0, o1); u = u01(o0); }
  else               { tf2x32(ku0, ku1, (uint32_t)b - BATCH / 2u, (uint32_t)b, o0, o1); u = u01(o1); }
  u = fmaxf(u, 0.0f);

  const float mean = dsum * (1.0f / 48.0f);
  const float var  = dsq * (1.0f / 48.0f) - mean * mean;
  const float vstd = sqrtf(fmaxf(var, 0.0f));
  const float mat  = ((float)mi + u) * (1.0f / 49.0f);

  out[b * 3 + 0] = (mx   - 0.38f) * (1.0f / 0.14f);
  out[b * 3 + 1] = (mat  - 0.21f) * (1.0f / 0.12f);
  out[b * 3 + 2] = (vstd - 0.14f) * (1.0f / 0.03f);
}

// Observation update: nan_to_num -> clip(1e-5) -> max/argmax + log-diff stats
__device__ __forceinline__ void save_update(float iv, int t,
                                            float& mx, int& mi,
                                            float& prev_lx, float& dsum, float& dsq) {
  float x = iv;
  if (!isfinite(x)) x = 0.0f;
  x = fmaxf(x, 1e-5f);
  if (x > mx) { mx = x; mi = t; }
  const float lx = logf(x);
  if (t > 0) { const float d = lx - prev_lx; dsum += d; dsq += d * d; }
  prev_lx = lx;
}

// ---------------- phase 2 (WS path): LDS double-buffered integration ----------------
__global__ void sir_sim_lds_kernel(const float* __restrict__ cond,
                                   const uint32_t* __restrict__ keys,
                                   const float* __restrict__ noise,
                                   float* __restrict__ out) {
  __shared__ float tile[2][CHUNK_ROWS * TPB];   // 2 x 20KB

  const int tid    = threadIdx.x;
  const int b_base = blockIdx.x * TPB;
  const int b      = b_base + tid;

  const uint32_t lb0 = (uint32_t)(uintptr_t)(&tile[0][0]);
  const uint32_t lb1 = (uint32_t)(uintptr_t)(&tile[1][0]);

  // Issue one chunk's async copies: 20 rows x 256 floats, b128 per lane.
  auto issue_chunk = [&](int chunk, uint32_t lds_base) {
#pragma unroll
    for (int it = 0; it < 5; ++it) {
      const int idx = it * TPB + tid;        // 0..1279
      const int r   = idx >> 6;              // 0..19
      const int c4  = idx & 63;              // 0..63
      const float* g = noise + (size_t)(chunk * CHUNK_ROWS + r) * BATCH + b_base + c4 * 4;
      async_load_b128(lds_base + (uint32_t)(r * (TPB * 4) + c4 * 16), g);
    }
  };

  const uint32_t ku0 = keys[2], ku1 = keys[3];

  const float pinf = fminf(fmaxf(cond[b * 4 + 0], 1e-7f), 1.0f - 1e-7f);
  const float prec = fminf(fmaxf(cond[b * 4 + 1], 1e-7f), 1.0f - 1e-7f);
  const float pmr  = fminf(fmaxf(cond[b * 4 + 2], 1e-7f), 1.0f - 1e-7f);
  const float pvol = fminf(fmaxf(cond[b * 4 + 3], 1e-7f), 1.0f - 1e-7f);

  const float r0m  = pinf / prec;
  const float dtmr = DTC * pmr;

  float s = 0.99f, i = 0.01f, r0 = r0m;
  float mx = -1.0f; int mi = 0;
  float prev_lx = 0.0f, dsum = 0.0f, dsq = 0.0f;

  issue_chunk(0, lb0);

  for (int c = 0; c < NUM_CHUNKS; ++c) {
    const uint32_t cur = (c & 1) ? lb1 : lb0;
    if (c + 1 < NUM_CHUNKS) {
      issue_chunk(c + 1, ((c + 1) & 1) ? lb1 : lb0);
      wait_async_le5();                  // chunk c complete (async loads are in-order)
    } else {
      wait_async_le0();
    }
    __syncthreads();                     // chunk c visible to all waves

    const float* lp = (const float*)(uintptr_t)0;  // (unused; LDS read via array below)
    (void)lp;
#pragma unroll
    for (int r = 0; r < CHUNK_ROWS; ++r) {
      const float dw  = tile[c & 1][r * TPB + tid];
      const float nf  = (r0 * prec) * s;
      const float nr  = prec * i;
      const float s_n = s - DTC * nf;
      const float i_n = i + DTC * (nf - nr);
      const float r0n = r0 + dtmr * (r0m - r0) + (sqrtf(fabsf(r0)) * pvol) * dw;
      s = s_n; i = i_n; r0 = r0n;
    }
    if ((c % CHUNKS_PER_SAVE) == (CHUNKS_PER_SAVE - 1)) {
      save_update(i, c / CHUNKS_PER_SAVE, mx, mi, prev_lx, dsum, dsq);
    }
    __syncthreads();                     // all reads done before buffer reuse
  }

  write_summary(out, b, ku0, ku1, mx, mi, dsum, dsq);
}

// ---------------- phase 2 (fallback): fused counter-based regeneration ----------------
__global__ void sir_sim_fused_kernel(const float* __restrict__ cond,
                                     const int* __restrict__ seedp,
                                     float* __restrict__ out) {
  const int b = blockIdx.x * blockDim.x + threadIdx.x;
  if (b >= BATCH) return;

  long long sd = (long long)(*seedp);
  uint32_t k0 = (uint32_t)(((unsigned long long)sd) >> 32);
  uint32_t k1 = (uint32_t)((unsigned long long)sd);
  uint32_t a0, a1, c0, c1;
  tf2x32(k0, k1, 0u, 2u, a0, a1);
  tf2x32(k0, k1, 1u, 3u, c0, c1);
  const uint32_t kn0 = a0, kn1 = c0, ku0 = a1, ku1 = c1;

  const float pinf = fminf(fmaxf(cond[b * 4 + 0], 1e-7f), 1.0f - 1e-7f);
  const float prec = fminf(fmaxf(cond[b * 4 + 1], 1e-7f), 1.0f - 1e-7f);
  const float pmr  = fminf(fmaxf(cond[b * 4 + 2], 1e-7f), 1.0f - 1e-7f);
  const float pvol = fminf(fmaxf(cond[b * 4 + 3], 1e-7f), 1.0f - 1e-7f);

  const float r0m  = pinf / prec;
  const float dtmr = DTC * pmr;

  float s = 0.99f, i = 0.01f, r0 = r0m;
  float mx = -1.0f; int mi = 0;
  float prev_lx = 0.0f, dsum = 0.0f, dsq = 0.0f;

  for (int t = 0; t < TIME_STEPS; ++t) {
#pragma unroll 4
    for (int k = 0; k < STEPS_PER_SAVE; ++k) {
      const int st = t * STEPS_PER_SAVE + k;
      const uint32_t idx = (uint32_t)st * (uint32_t)BATCH + (uint32_t)b;
      uint32_t o0, o1; float dw;
      if (st < HALF_ST) { tf2x32(kn0, kn1, idx, idx + NHALF, o0, o1); dw = normal_scaled(o0); }
      else              { tf2x32(kn0, kn1, idx - NHALF, idx, o0, o1); dw = normal_scaled(o1); }
      const float nf  = (r0 * prec) * s;
      const float nr  = prec * i;
      const float s_n = s - DTC * nf;
      const float i_n = i + DTC * (nf - nr);
      const float r0n = r0 + dtmr * (r0m - r0) + (sqrtf(fabsf(r0)) * pvol) * dw;
      s = s_n; i = i_n; r0 = r0n;
    }
    save_update(i, t, mx, mi, prev_lx, dsum, dsq);
  }

  write_summary(out, b, ku0, ku1, mx, mi, dsum, dsq);
}

extern "C" void kernel_launch(void* const* d_in, const int* in_sizes, int n_in,
                              void* d_out, int out_size, void* d_ws, size_t ws_size,
                              hipStream_t stream) {
  (void)in_sizes; (void)n_in; (void)out_size;
  const float* cond  = (const float*)d_in[0];
  const int*   seedp = (const int*)d_in[1];
  float*       out   = (float*)d_out;

  const size_t noise_bytes = (size_t)NOISE_N * sizeof(float);
  const size_t need        = noise_bytes + 4 * sizeof(uint32_t);

  if (d_ws != nullptr && ws_size >= need) {
    float*    noise = (float*)d_ws;
    uint32_t* keys  = (uint32_t*)((char*)d_ws + noise_bytes);
    derive_keys_kernel<<<1, 32, 0, stream>>>(seedp, keys);
    gen_noise_kernel<<<(NHALF + TPB - 1u) / TPB, TPB, 0, stream>>>(noise, keys);
    sir_sim_lds_kernel<<<BATCH / TPB, TPB, 0, stream>>>(cond, keys, noise, out);
  } else {
    sir_sim_fused_kernel<<<BATCH / TPB, TPB, 0, stream>>>(cond, seedp, out);
  }
}